// PressureToSigmaInterpolator_33320356282814
// MI455X (gfx1250) — compile-verified
//
#include <hip/hip_runtime.h>

// Pressure -> hybrid-sigma vertical interpolation (ERA5-style).
// Shapes (float32):
//   vars_era5 : (5, NLAT, NLON, NLEV)
//   ps        : (NLAT, NLON)
//   a_k, b_k  : (NZ)
//   plev      : (NLEV)  ascending (Pa)
// Output: (6, NLAT, NLON, NZ)  = 5 interpolated vars + p_model.

#define NLAT 361
#define NLON 720
#define NLEV 37
#define NZ   32
#define NPTS (NLAT * NLON)
#define EPSF 1e-8f
#define PTS_PER_BLOCK 8   // 8 wave32s per 256-thread block, one point per wave

__global__ __launch_bounds__(256) void interp_p2s_kernel(
    const float* __restrict__ vars,  // (5, NPTS, NLEV)
    const float* __restrict__ ps,    // (NPTS)
    const float* __restrict__ a_k,   // (NZ)
    const float* __restrict__ b_k,   // (NZ)
    const float* __restrict__ plev,  // (NLEV)
    float* __restrict__ out)         // (6, NPTS, NZ)
{
    // LDS tables: raw levels (for the linear-space search), log2 levels,
    // and reciprocal log2-interval widths (replaces the divide in t).
    __shared__ float s_lev[NLEV];
    __shared__ float s_log[NLEV];
    __shared__ float s_inv[NLEV];

    const int tid = threadIdx.x;
    if (tid < NLEV) {
        float lv = plev[tid];
        s_lev[tid] = lv;
        s_log[tid] = __log2f(lv + EPSF);
        if (tid >= 1) {
            float lv0 = plev[tid - 1];
            // fast hardware reciprocal: feeds an interp weight, ~1 ulp is fine,
            // and this runs only once per block for 36 lanes.
            s_inv[tid] = __builtin_amdgcn_rcpf(__log2f(lv + EPSF) -
                                               __log2f(lv0 + EPSF));
        }
    }
    __syncthreads();

    const int wave = tid >> 5;        // wave32: 8 waves per block
    const int lane = tid & 31;        // lane = sigma level k (NZ == 32)
    const int p = blockIdx.x * PTS_PER_BLOCK + wave;
    if (p >= NPTS) return;

    // Prefetch the five 148-byte gather rows for this point (global_prefetch_b8).
    #pragma unroll
    for (int v = 0; v < 5; ++v) {
        __builtin_prefetch(vars + ((size_t)v * NPTS + p) * NLEV, 0, 1);
    }

    const float psv = ps[p];                            // uniform within wave
    const float pm  = fmaf(b_k[lane], psv, a_k[lane]);  // p_model
    const float lpm = __log2f(pm + EPSF);

    // searchsorted(log_lev, log_pm) == count(lev < pm)   (log is monotone)
    // n_valid = count(lev <= ps)
    int idx = 0, nv = 0;
    #pragma unroll
    for (int l = 0; l < NLEV; ++l) {
        float lv = s_lev[l];          // uniform LDS broadcast read
        idx += (lv <  pm)  ? 1 : 0;
        nv  += (lv <= psv) ? 1 : 0;
    }

    int hi = nv - 1; if (hi < 1) hi = 1;
    if (idx < 1)  idx = 1;
    if (idx > hi) idx = hi;

    const float x0 = s_log[idx - 1];
    const float t  = (lpm - x0) * s_inv[idx];   // base-2 logs: ln2 cancels
    const bool valid = (nv >= 2);

    const size_t rbase = (size_t)p * NLEV;
    const size_t obase = (size_t)p * NZ + lane;

    // Issue all 10 gathers first so the loads overlap (all inside one cached row).
    float y0[5], y1[5];
    #pragma unroll
    for (int v = 0; v < 5; ++v) {
        const float* row = vars + (size_t)v * NPTS * NLEV + rbase;
        y0[v] = row[idx - 1];
        y1[v] = row[idx];
    }

    // Streaming output: non-temporal stores (TH=NT) keep WGP$/L2 capacity
    // for the gather rows, which *do* have short-term reuse.
    #pragma unroll
    for (int v = 0; v < 5; ++v) {
        float r = valid ? fmaf(t, y1[v] - y0[v], y0[v]) : 0.0f;
        __builtin_nontemporal_store(r, &out[(size_t)v * NPTS * NZ + obase]);
    }
    // p_model channel is written unconditionally (matches reference concat).
    __builtin_nontemporal_store(pm, &out[(size_t)5 * NPTS * NZ + obase]);
}

extern "C" void kernel_launch(void* const* d_in, const int* in_sizes, int n_in,
                              void* d_out, int out_size, void* d_ws, size_t ws_size,
                              hipStream_t stream) {
    const float* vars = (const float*)d_in[0];   // (5, NLAT, NLON, NLEV)
    const float* psp  = (const float*)d_in[1];   // (NLAT, NLON)
    const float* ak   = (const float*)d_in[2];   // (NZ)
    const float* bk   = (const float*)d_in[3];   // (NZ)
    const float* plev = (const float*)d_in[4];   // (NLEV)
    float* out = (float*)d_out;                  // (6, NLAT, NLON, NZ)

    const int blocks = (NPTS + PTS_PER_BLOCK - 1) / PTS_PER_BLOCK;
    interp_p2s_kernel<<<blocks, 256, 0, stream>>>(vars, psp, ak, bk, plev, out);
}